// Convolution1D_Locally_Connected_72868415144016
// MI455X (gfx1250) — compile-verified
//
#include <hip/hip_runtime.h>
#include <stdint.h>

// Problem constants (from reference setup_inputs)
enum : int { B = 8, F = 64, L = 131072, KTAPS = 9 };
enum : int { TILE = 4096, HALO = 4, TPB = 256, PER_THREAD = TILE / TPB }; // 16

typedef float v4f __attribute__((ext_vector_type(4)));

__global__ __launch_bounds__(TPB) void dwconv1d_async_kernel(
    const float* __restrict__ x,     // (B, F, L)
    const float* __restrict__ w,     // (F, 1, K)
    float* __restrict__ out)         // (F, B, 1, L)
{
    __shared__ __align__(16) float s[TILE + 2 * HALO];

    const int tid  = threadIdx.x;
    const int tile = blockIdx.x;          // 0..L/TILE-1
    const int row  = blockIdx.y;          // row = b*F + f  (x layout)
    const int f    = row & (F - 1);
    const int b    = row >> 6;            // row / F, F = 64
    const long t0  = (long)tile * TILE;

    const float* __restrict__ xrow = x + (size_t)row * L + t0;                 // in-row base
    float*       __restrict__ orow = out + ((size_t)f * B + b) * (size_t)L + t0;

    // ---- bulk tile: async global -> LDS, 16B per lane per issue (always in-bounds) ----
    {
        const uint32_t lds0 = (uint32_t)(uintptr_t)&s[HALO];
        #pragma unroll
        for (int i = 0; i < TILE / 4 / TPB; ++i) {          // 4 issues per thread
            const uint32_t g    = (uint32_t)(tid + i * TPB); // float4 group index
            const uint32_t loff = lds0 + g * 16u;            // LDS byte address
            const uint32_t goff = g * 16u;                   // global byte offset (GVS)
            asm volatile("global_load_async_to_lds_b128 %0, %1, %2"
                         :: "v"(loff), "v"(goff), "s"(xrow)
                         : "memory");
        }
    }

    // ---- halo (8 floats, guarded; zeros give exact SAME padding at row edges) ----
    if (tid < 2 * HALO) {
        const bool left = tid < HALO;
        const long gi   = left ? (t0 - HALO + tid) : (t0 + TILE + (tid - HALO));
        const int  si   = left ? tid : (TILE + HALO + (tid - HALO));
        float v = 0.0f;
        if (gi >= 0 && gi < (long)L) v = x[(size_t)row * L + (size_t)gi];
        s[si] = v;
    }

    // each wave drains its own async copies, then workgroup barrier
    asm volatile("s_wait_asynccnt 0x0" ::: "memory");
    __syncthreads();

    // ---- taps (uniform across block -> scalar loads) ----
    const float* __restrict__ wf = w + (size_t)f * KTAPS;
    const float w0 = wf[0], w1 = wf[1], w2 = wf[2], w3 = wf[3], w4 = wf[4],
                w5 = wf[5], w6 = wf[6], w7 = wf[7], w8 = wf[8];

    // ---- pull 24-float window from LDS (6 x ds_load_b128) ----
    const int base = tid * PER_THREAD;                 // 16-float aligned
    float r[PER_THREAD + 2 * HALO];
    {
        const v4f* __restrict__ sp = (const v4f*)&s[base];
        #pragma unroll
        for (int i = 0; i < (PER_THREAD + 2 * HALO) / 4; ++i) {
            v4f v = sp[i];
            r[4 * i + 0] = v.x; r[4 * i + 1] = v.y;
            r[4 * i + 2] = v.z; r[4 * i + 3] = v.w;
        }
    }

    // ---- 9-tap FIR, register sliding window: out[l] = sum_k w[k] * x[l + k - 4] ----
    float acc[PER_THREAD];
    #pragma unroll
    for (int i = 0; i < PER_THREAD; ++i) {
        float a =        r[i + 0] * w0;
        a = fmaf(r[i + 1], w1, a);
        a = fmaf(r[i + 2], w2, a);
        a = fmaf(r[i + 3], w3, a);
        a = fmaf(r[i + 4], w4, a);
        a = fmaf(r[i + 5], w5, a);
        a = fmaf(r[i + 6], w6, a);
        a = fmaf(r[i + 7], w7, a);
        a = fmaf(r[i + 8], w8, a);
        acc[i] = a;
    }

    // ---- non-temporal b128 stores (output is write-once, never re-read) ----
    {
        v4f* __restrict__ op = (v4f*)(orow + base);
        #pragma unroll
        for (int i = 0; i < PER_THREAD / 4; ++i) {
            v4f v;
            v.x = acc[4 * i + 0]; v.y = acc[4 * i + 1];
            v.z = acc[4 * i + 2]; v.w = acc[4 * i + 3];
            __builtin_nontemporal_store(v, op + i);
        }
    }
}

extern "C" void kernel_launch(void* const* d_in, const int* in_sizes, int n_in,
                              void* d_out, int out_size, void* d_ws, size_t ws_size,
                              hipStream_t stream) {
    (void)in_sizes; (void)n_in; (void)d_ws; (void)ws_size; (void)out_size;
    const float* x = (const float*)d_in[0];   // (B, F, L) fp32
    const float* w = (const float*)d_in[1];   // (F, 1, K) fp32
    float* out     = (float*)d_out;           // (F, B, 1, L) fp32

    dim3 grid(L / TILE, B * F, 1);            // (32, 512)
    dwconv1d_async_kernel<<<grid, TPB, 0, stream>>>(x, w, out);
}